// OMGNN_RNN_88046829568213
// MI455X (gfx1250) — compile-verified
//
#include <hip/hip_runtime.h>
#include <hip/hip_bf16.h>

// CDNA5 / gfx1250, wave32. All GEMMs via V_WMMA_F32_16X16X4_F32 (f32-exact;
// workload is HBM-bound at ~17 FLOP/B so f32 precision costs nothing).

typedef __attribute__((ext_vector_type(2))) float v2f;
typedef __attribute__((ext_vector_type(8))) float v8f;

#define HID 160        // hidden size (10 tiles of 16)
#define NTILE 10
#define ROWS 128       // edge/node rows per block (8 waves x 16)
#define TB 256

static __device__ __forceinline__ v8f wmma4(v2f a, v2f b, v8f c) {
  // 8 args: (neg_a, A, neg_b, B, c_mod, C, reuse_a, reuse_b)
  return __builtin_amdgcn_wmma_f32_16x16x4_f32(false, a, false, b, (short)0, c,
                                               false, false);
}

// Wave-level 16xHID GEMM over K (multiple of 4), A in LDS row-major [16 x K],
// B in LDS pair layout ldsB[p*HID+n] = {B[2p][n], B[2p+1][n]}.
// Software-pipelined: next step's A + 10 B fragments prefetched into regs
// while the current 10 WMMAs issue (low occupancy: ~200KB LDS/block).
template <int K>
static __device__ __forceinline__ void wave_gemm(const float* __restrict__ aRow,
                                                 const v2f* __restrict__ ldsB,
                                                 int m, int kh,
                                                 v8f acc[NTILE]) {
  constexpr int K4 = K / 4;
  const float* aPtr = aRow + kh * 2;       // per-lane K offset
  const v2f* bBase = &ldsB[kh * HID + m];  // per-lane pair base

  v2f a_cur = *(const v2f*)aPtr;
  v2f b_cur[NTILE];
#pragma unroll
  for (int t = 0; t < NTILE; ++t) b_cur[t] = bBase[t * 16];

#pragma unroll
  for (int kk = 0; kk < K4; ++kk) {
    v2f a_nxt{};
    v2f b_nxt[NTILE];
    if (kk + 1 < K4) {
      a_nxt = *(const v2f*)(aPtr + (kk + 1) * 4);
      const v2f* bp = bBase + (kk + 1) * 2 * HID;
#pragma unroll
      for (int t = 0; t < NTILE; ++t) b_nxt[t] = bp[t * 16];
    }
#pragma unroll
    for (int t = 0; t < NTILE; ++t) acc[t] = wmma4(a_cur, b_cur[t], acc[t]);
    a_cur = a_nxt;
#pragma unroll
    for (int t = 0; t < NTILE; ++t) b_cur[t] = b_nxt[t];
  }
}

// ---------------------------------------------------------------------------
// Kernel 1: H0 = relu([x[src] ; edge_attr ; pad2] @ Wi^T + bi); also H = H0
// K = 176 (174 padded to /4). LDS: sidx(512B) + A(128x176 f32) + B(88x160 v2f)
// ---------------------------------------------------------------------------
#define KA 176
#define SMEM_H0 (512 + ROWS * KA * 4 + (KA / 2) * HID * 8)

__global__ __launch_bounds__(TB) void k_h0(const float* __restrict__ x,
                                           const float* __restrict__ ea,
                                           const int* __restrict__ src,
                                           const float* __restrict__ Wi,
                                           const float* __restrict__ bi,
                                           float* __restrict__ H0,
                                           float* __restrict__ H, int E) {
  extern __shared__ char smem[];
  int* sidx = (int*)smem;
  float* ldsA = (float*)(smem + 512);
  v2f* ldsB = (v2f*)(smem + 512 + ROWS * KA * 4);
  const int tid = threadIdx.x;
  const int base = blockIdx.x * ROWS;

  if (tid < ROWS) {
    int e = base + tid;
    sidx[tid] = (e < E) ? src[e] : 0;
  }
  // B[k][n] = Wi[n*174 + k]; 174 even so pairs never straddle the pad
  for (int i = tid; i < (KA / 2) * HID; i += TB) {
    int p = i / HID, n = i - p * HID;
    v2f b = (v2f)(0.0f);
    if (2 * p < 174) b = *(const v2f*)&Wi[n * 174 + 2 * p];
    ldsB[p * HID + n] = b;
  }
  __syncthreads();
  // A[r][c] = c<160 ? x[src[e]][c] : (c<174 ? edge_attr[e][c-160] : 0)
  for (int i = tid; i < ROWS * KA; i += TB) {
    int r = i / KA, c = i - r * KA;
    int e = base + r;
    float v = 0.0f;
    if (e < E) {
      if (c < 160)       v = x[(size_t)sidx[r] * 160 + c];
      else if (c < 174)  v = ea[(size_t)e * 14 + (c - 160)];
    }
    ldsA[i] = v;
  }
  __syncthreads();

  const int wave = tid >> 5, lane = tid & 31;
  const int m = lane & 15, kh = lane >> 4;
  v8f acc[NTILE] = {};
  wave_gemm<KA>(&ldsA[(wave * 16 + m) * KA], ldsB, m, kh, acc);

#pragma unroll
  for (int t = 0; t < NTILE; ++t) {
    int n = t * 16 + m;
    float bias = bi[n];
#pragma unroll
    for (int r = 0; r < 8; ++r) {
      int e = base + wave * 16 + r + kh * 8;
      if (e < E) {
        float v = fmaxf(acc[t][r] + bias, 0.0f);
        H0[(size_t)e * HID + n] = v;
        H[(size_t)e * HID + n] = v;
      }
    }
  }
}

// ---------------------------------------------------------------------------
// Kernel 2: Hout = relu(H0 + (Magg[src] - Hin[rev]) @ Wh^T + bh)   (K = 160)
// Message formation fused into A staging (never materialize M in HBM).
// ---------------------------------------------------------------------------
#define KB 160
#define SMEM_UP (1024 + ROWS * KB * 4 + (KB / 2) * HID * 8)

__global__ __launch_bounds__(TB) void k_update(
    const float* __restrict__ Magg, const float* __restrict__ Hin,
    const float* __restrict__ H0, const int* __restrict__ src,
    const int* __restrict__ rev, const float* __restrict__ Wh,
    const float* __restrict__ bh, float* __restrict__ Hout, int E) {
  extern __shared__ char smem[];
  int* sidx = (int*)smem;
  int* ridx = (int*)(smem + 512);
  float* ldsA = (float*)(smem + 1024);
  v2f* ldsB = (v2f*)(smem + 1024 + ROWS * KB * 4);
  const int tid = threadIdx.x;
  const int base = blockIdx.x * ROWS;

  if (tid < ROWS) {
    int e = base + tid;
    sidx[tid] = (e < E) ? src[e] : 0;
    ridx[tid] = (e < E) ? rev[e] : 0;
  }
  // B[k][n] = Wh[n*160 + k]; pairs contiguous in Wh rows
  for (int i = tid; i < (KB / 2) * HID; i += TB) {
    int p = i / HID, n = i - p * HID;
    ldsB[p * HID + n] = *(const v2f*)&Wh[n * KB + 2 * p];
  }
  __syncthreads();
  for (int i = tid; i < ROWS * KB; i += TB) {
    int r = i / KB, c = i - r * KB;
    int e = base + r;
    float v = 0.0f;
    if (e < E)
      v = Magg[(size_t)sidx[r] * HID + c] - Hin[(size_t)ridx[r] * HID + c];
    ldsA[i] = v;
  }
  __syncthreads();

  const int wave = tid >> 5, lane = tid & 31;
  const int m = lane & 15, kh = lane >> 4;
  v8f acc[NTILE] = {};
  wave_gemm<KB>(&ldsA[(wave * 16 + m) * KB], ldsB, m, kh, acc);

#pragma unroll
  for (int t = 0; t < NTILE; ++t) {
    int n = t * 16 + m;
    float bias = bh[n];
#pragma unroll
    for (int r = 0; r < 8; ++r) {
      int e = base + wave * 16 + r + kh * 8;
      if (e < E) {
        float v = acc[t][r] + bias + H0[(size_t)e * HID + n];
        Hout[(size_t)e * HID + n] = fmaxf(v, 0.0f);
      }
    }
  }
}

// ---------------------------------------------------------------------------
// Kernel 3: out = relu([x ; M] @ Wo^T + bo), K = 320 processed as two halves
// reusing the same LDS (A half + B half restaged per phase, acc persists).
// ---------------------------------------------------------------------------
#define SMEM_OUT (ROWS * KB * 4 + (KB / 2) * HID * 8)

__global__ __launch_bounds__(TB) void k_out(const float* __restrict__ x,
                                            const float* __restrict__ M,
                                            const float* __restrict__ Wo,
                                            const float* __restrict__ bo,
                                            float* __restrict__ out, int N) {
  extern __shared__ char smem[];
  float* ldsA = (float*)smem;
  v2f* ldsB = (v2f*)(smem + ROWS * KB * 4);
  const int tid = threadIdx.x;
  const int base = blockIdx.x * ROWS;
  const int wave = tid >> 5, lane = tid & 31;
  const int m = lane & 15, kh = lane >> 4;
  v8f acc[NTILE] = {};

  for (int half = 0; half < 2; ++half) {
    __syncthreads();  // previous phase's LDS reads done before restaging
    const float* srcm = half ? M : x;
    for (int i = tid; i < (KB / 2) * HID; i += TB) {
      int p = i / HID, n = i - p * HID;
      ldsB[p * HID + n] = *(const v2f*)&Wo[n * 320 + half * 160 + 2 * p];
    }
    for (int i = tid; i < ROWS * KB; i += TB) {
      int r = i / KB, c = i - r * KB;
      int node = base + r;
      ldsA[i] = (node < N) ? srcm[(size_t)node * HID + c] : 0.0f;
    }
    __syncthreads();
    wave_gemm<KB>(&ldsA[(wave * 16 + m) * KB], ldsB, m, kh, acc);
  }
#pragma unroll
  for (int t = 0; t < NTILE; ++t) {
    int n = t * 16 + m;
    float bias = bo[n];
#pragma unroll
    for (int r = 0; r < 8; ++r) {
      int node = base + wave * 16 + r + kh * 8;
      if (node < N)
        out[(size_t)node * HID + n] = fmaxf(acc[t][r] + bias, 0.0f);
    }
  }
}

// ---------------------------------------------------------------------------
// Helpers: zero-fill, scatter-add (segment_sum), zero-row fixup
// ---------------------------------------------------------------------------
__global__ void k_zero4(float4* __restrict__ p, int n4) {
  int i = blockIdx.x * blockDim.x + threadIdx.x;
  if (i < n4) p[i] = make_float4(0.f, 0.f, 0.f, 0.f);
}

__global__ void k_scatter(const float* __restrict__ H,
                          const int* __restrict__ dst,
                          float* __restrict__ Magg, int E) {
  int tid = blockIdx.x * blockDim.x + threadIdx.x;
  int e = tid / 40;
  if (e >= E) return;
  int c = (tid - e * 40) * 4;
  float4 h = *(const float4*)&H[(size_t)e * HID + c];
  float* o = &Magg[(size_t)dst[e] * HID + c];
  unsafeAtomicAdd(o + 0, h.x);
  unsafeAtomicAdd(o + 1, h.y);
  unsafeAtomicAdd(o + 2, h.z);
  unsafeAtomicAdd(o + 3, h.w);
}

// one wave per node: if rowsum(M[i]) == 0, M[i] = x[i]
__global__ void k_fixup(const float* __restrict__ x, float* __restrict__ M,
                        int N) {
  int node = (blockIdx.x * blockDim.x + threadIdx.x) >> 5;
  int lane = threadIdx.x & 31;
  if (node >= N) return;
  float s = 0.f;
#pragma unroll
  for (int j = 0; j < 5; ++j) s += M[(size_t)node * HID + j * 32 + lane];
#pragma unroll
  for (int off = 16; off; off >>= 1) s += __shfl_down(s, off, 32);
  s = __shfl(s, 0, 32);
  if (s == 0.0f) {
#pragma unroll
    for (int j = 0; j < 5; ++j)
      M[(size_t)node * HID + j * 32 + lane] =
          x[(size_t)node * HID + j * 32 + lane];
  }
}

// ---------------------------------------------------------------------------
extern "C" void kernel_launch(void* const* d_in, const int* in_sizes, int n_in,
                              void* d_out, int out_size, void* d_ws,
                              size_t ws_size, hipStream_t stream) {
  const float* x  = (const float*)d_in[0];
  const float* ea = (const float*)d_in[1];
  const int* eidx = (const int*)d_in[2];   // [2,E]: row0=src, row1=dst
  const int* rev  = (const int*)d_in[3];
  const float* Wi = (const float*)d_in[4];
  const float* bi = (const float*)d_in[5];
  const float* Wh = (const float*)d_in[6];
  const float* bh = (const float*)d_in[7];
  const float* Wo = (const float*)d_in[8];
  const float* bo = (const float*)d_in[9];
  float* out = (float*)d_out;

  const int E = in_sizes[3];
  const int N = in_sizes[0] / HID;
  const int* srcI = eidx;
  const int* dstI = eidx + E;

  float* H0 = (float*)d_ws;
  float* Ha = H0 + (size_t)E * HID;
  float* Hb = Ha + (size_t)E * HID;
  float* Magg = Hb + (size_t)E * HID;

  const int blkE = (E + ROWS - 1) / ROWS;
  const int blkN = (N + ROWS - 1) / ROWS;
  const int blkScat = (E * 40 + TB - 1) / TB;
  const int blkZero = (N * HID / 4 + TB - 1) / TB;
  const int blkFix = (N * 32 + TB - 1) / TB;

  // H = H0 = relu(Wi([x_src ; e]))
  k_h0<<<blkE, TB, SMEM_H0, stream>>>(x, ea, srcI, Wi, bi, H0, Ha, E);

  // iteration 1: Ha -> Hb
  k_zero4<<<blkZero, TB, 0, stream>>>((float4*)Magg, N * HID / 4);
  k_scatter<<<blkScat, TB, 0, stream>>>(Ha, dstI, Magg, E);
  k_update<<<blkE, TB, SMEM_UP, stream>>>(Magg, Ha, H0, srcI, rev, Wh, bh, Hb,
                                          E);
  // iteration 2: Hb -> Ha
  k_zero4<<<blkZero, TB, 0, stream>>>((float4*)Magg, N * HID / 4);
  k_scatter<<<blkScat, TB, 0, stream>>>(Hb, dstI, Magg, E);
  k_update<<<blkE, TB, SMEM_UP, stream>>>(Magg, Hb, H0, srcI, rev, Wh, bh, Ha,
                                          E);
  // final aggregation + fixup + readout
  k_zero4<<<blkZero, TB, 0, stream>>>((float4*)Magg, N * HID / 4);
  k_scatter<<<blkScat, TB, 0, stream>>>(Ha, dstI, Magg, E);
  k_fixup<<<blkFix, TB, 0, stream>>>(x, Magg, N);
  k_out<<<blkN, TB, SMEM_OUT, stream>>>(x, Magg, Wo, bo, out, N);
}